// GMM_Layer_75874892251650
// MI455X (gfx1250) — compile-verified
//
#include <hip/hip_runtime.h>
#include <math.h>

// ---------------------------------------------------------------------------
// GMM layer: logprobs[b,k] = -0.5*sum_c (x-mu)^2*sigma^2 + sum_c log sigma
//                            + CONST + log_softmax(pis)[k]
// Factored as fp32 WMMA GEMM:  A' = [x^2 | x]  (1024 x 1024)
//                              B'' = [-0.5*sigma^2 ; mu*sigma^2] (1024 x 256)
//                              out = A'.B'' + term[k]
// ---------------------------------------------------------------------------

typedef __attribute__((ext_vector_type(2))) float v2f;
typedef __attribute__((ext_vector_type(8))) float v8f;

#define Bn 1024
#define Kn 256
#define Cn 512
// -0.5 * C * log(2*pi)
#define GMM_CONST (-256.0f * 1.8378770664093453f)

// ---------------------------------------------------------------------------
// K1: logsumexp over the 256 mixture logits -> ws[0]
// ---------------------------------------------------------------------------
__global__ void __launch_bounds__(256)
lse_kernel(const float* __restrict__ pis, float* __restrict__ lse_out) {
    __shared__ float sm[256];
    const int t = threadIdx.x;
    const float p = pis[t];
    sm[t] = p;
    __syncthreads();
    for (int s = 128; s > 0; s >>= 1) {
        if (t < s) sm[t] = fmaxf(sm[t], sm[t + s]);
        __syncthreads();
    }
    const float m = sm[0];
    __syncthreads();
    sm[t] = expf(p - m);
    __syncthreads();
    for (int s = 128; s > 0; s >>= 1) {
        if (t < s) sm[t] += sm[t + s];
        __syncthreads();
    }
    if (t == 0) lse_out[0] = m + logf(sm[0]);
}

// ---------------------------------------------------------------------------
// K2: per-component scalar term:
//   term[k] = sum_c log(sigma) + CONST - 0.5*sum_c mu^2*sigma^2 + (pi[k]-lse)
// One block per k, 256 threads, 2 channels per thread.
// ---------------------------------------------------------------------------
__global__ void __launch_bounds__(256)
term_kernel(const float* __restrict__ pis, const float* __restrict__ mus,
            const float* __restrict__ sigmas, const float* __restrict__ lse_in,
            float* __restrict__ term) {
    __shared__ float sld[256];
    __shared__ float sct[256];
    const int k = blockIdx.x;
    const int t = threadIdx.x;
    const float* sg = sigmas + (size_t)k * Cn;
    const float* mu = mus + (size_t)k * Cn;
    float ld = 0.0f, ct = 0.0f;
    for (int c = t; c < Cn; c += 256) {
        const float s = sg[c];
        const float m = mu[c];
        const float s2 = s * s;
        ld += logf(s);
        ct += m * m * s2;
    }
    sld[t] = ld;
    sct[t] = ct;
    __syncthreads();
    for (int s = 128; s > 0; s >>= 1) {
        if (t < s) { sld[t] += sld[t + s]; sct[t] += sct[t + s]; }
        __syncthreads();
    }
    if (t == 0)
        term[k] = sld[0] + GMM_CONST - 0.5f * sct[0] + (pis[k] - lse_in[0]);
}

// ---------------------------------------------------------------------------
// K3: WMMA GEMM. One 16x16 output tile per wave32, K-dim = 1024 in steps of 4
// via V_WMMA_F32_16X16X4_F32 (full fp32 precision).
//
// Per-lane fragment layout (ISA 7.12.2, 32-bit A 16x4):
//   lanes 0-15 : a = {A[M=ln][K=kk+0], A[M=ln][K=kk+1]}
//   lanes16-31 : a = {A[M=ln][K=kk+2], A[M=ln][K=kk+3]}
// B (4x16) mirrored:  b = {B[K=kk+0+2*half][N=ln], B[K=kk+1+2*half][N=ln]}
// C/D: VGPR v, lanes 0-15 -> (M=v, N=ln); lanes 16-31 -> (M=8+v, N=ln).
// ---------------------------------------------------------------------------
__global__ void __launch_bounds__(256)
gmm_wmma_kernel(const float* __restrict__ x, const float* __restrict__ mus,
                const float* __restrict__ sigmas, const float* __restrict__ term,
                float* __restrict__ out) {
    const int lane  = threadIdx.x & 31;
    const int wave  = blockIdx.x * (blockDim.x >> 5) + (threadIdx.x >> 5);
    const int tileM = wave >> 4;   // 64 row tiles of 16
    const int tileN = wave & 15;   // 16 col tiles of 16
    const int row0  = tileM << 4;
    const int col0  = tileN << 4;
    const int half  = lane >> 4;   // 0: K pair {0,1}, 1: K pair {2,3}
    const int ln    = lane & 15;

    // A-side row for this lane (x), B-side column sources (sigma, mu).
    const float* __restrict__ xr = x      + (size_t)(row0 + ln) * Cn;
    const float* __restrict__ sr = sigmas + (size_t)(col0 + ln) * Cn;
    const float* __restrict__ mr = mus    + (size_t)(col0 + ln) * Cn;

    v8f acc = {};

    // ---- Pass 1: cc in [0,512):  A' = x^2,  B'' = -0.5 * sigma^2 ----------
#pragma unroll 4
    for (int kk = 0; kk < Cn; kk += 4) {
        const int ka = kk + (half << 1);          // even -> 8B aligned b64 load
        const v2f xv = *(const v2f*)(xr + ka);
        const v2f sv = *(const v2f*)(sr + ka);
        const v2f av = xv * xv;
        const v2f bv = (sv * sv) * -0.5f;
        acc = __builtin_amdgcn_wmma_f32_16x16x4_f32(
            /*neg_a=*/false, av, /*neg_b=*/false, bv,
            /*c_mod=*/(short)0, acc, /*reuse_a=*/false, /*reuse_b=*/false);
    }

    // ---- Pass 2: cc in [512,1024):  A' = x,  B'' = mu * sigma^2 -----------
#pragma unroll 4
    for (int kk = 0; kk < Cn; kk += 4) {
        const int ka = kk + (half << 1);
        const v2f av = *(const v2f*)(xr + ka);
        const v2f sv = *(const v2f*)(sr + ka);
        const v2f mv = *(const v2f*)(mr + ka);
        const v2f bv = mv * (sv * sv);
        acc = __builtin_amdgcn_wmma_f32_16x16x4_f32(
            false, av, false, bv, (short)0, acc, false, false);
    }

    // ---- Epilogue: add per-component scalar term, store tile --------------
    const float tk = term[col0 + ln];
#pragma unroll
    for (int v = 0; v < 8; ++v) {
        const int row = row0 + (half << 3) + v;
        out[(size_t)row * Kn + (col0 + ln)] = acc[v] + tk;
    }
}

// ---------------------------------------------------------------------------
extern "C" void kernel_launch(void* const* d_in, const int* in_sizes, int n_in,
                              void* d_out, int out_size, void* d_ws, size_t ws_size,
                              hipStream_t stream) {
    (void)in_sizes; (void)n_in; (void)out_size; (void)ws_size;
    const float* x      = (const float*)d_in[0];   // [1024, 512]
    const float* pis    = (const float*)d_in[1];   // [256]
    const float* mus    = (const float*)d_in[2];   // [256, 512]
    const float* sigmas = (const float*)d_in[3];   // [256, 512]
    float* out = (float*)d_out;                    // [1024, 256]

    float* wsf  = (float*)d_ws;
    float* lse  = wsf;        // 1 float
    float* term = wsf + 16;   // 256 floats (64B-offset for alignment)

    lse_kernel<<<1, 256, 0, stream>>>(pis, lse);
    term_kernel<<<Kn, 256, 0, stream>>>(pis, mus, sigmas, lse, term);
    // 1024 output tiles (64 x 16), 8 wave32 per block -> 128 blocks.
    gmm_wmma_kernel<<<128, 256, 0, stream>>>(x, mus, sigmas, term, out);
}